// JointLengthLoss_75093208203682
// MI455X (gfx1250) — compile-verified
//
#include <hip/hip_runtime.h>
#include <stdint.h>

// ---------------------------------------------------------------------------
// JointLengthLoss: mean over (B,20) of |‖pred_a-pred_b‖ - ‖gt_a-gt_b‖| / ‖gt_a-gt_b‖
// Memory-bound streaming reduction: 264 MB read -> ~11.3 us at 23.3 TB/s.
// Data path: CDNA5 async global->LDS copies (ASYNCcnt), tile = 128 samples.
// ---------------------------------------------------------------------------

#define TILE              128                    // samples per block
#define FPS               63                     // floats per sample (21*3)
#define TILE_FLOATS       (TILE * FPS)           // 8064
#define TILE_VEC16        (TILE_FLOATS / 4)      // 2016 x 16B chunks (exact)
#define NBONES            20

typedef int v4i __attribute__((ext_vector_type(4)));
typedef __attribute__((address_space(1))) v4i* gptr_v4i;
typedef __attribute__((address_space(3))) v4i* lptr_v4i;

// ----- CDNA5 async global->LDS helpers -------------------------------------
__device__ __forceinline__ void async_copy16(const float* gsrc, float* ldst) {
#if __has_builtin(__builtin_amdgcn_global_load_async_to_lds_b128)
  __builtin_amdgcn_global_load_async_to_lds_b128(
      (gptr_v4i)(uintptr_t)gsrc,
      (lptr_v4i)ldst,
      0, 0);
#else
  unsigned lds_off =
      (unsigned)(uintptr_t)(__attribute__((address_space(3))) void*)ldst;
  unsigned long long ga = (unsigned long long)(uintptr_t)gsrc;
  asm volatile("global_load_async_to_lds_b128 %0, %1, off"
               :: "v"(lds_off), "v"(ga) : "memory");
#endif
}

__device__ __forceinline__ void wait_async0() {
#if __has_builtin(__builtin_amdgcn_s_wait_asynccnt)
  __builtin_amdgcn_s_wait_asynccnt(0);
#else
  asm volatile("s_wait_asynccnt 0" ::: "memory");
#endif
}

// ----- Kernel 1: per-tile bone relative-error partial sums ------------------
__global__ __launch_bounds__(256) void
bone_tile_kernel(const float* __restrict__ pred,
                 const float* __restrict__ gt,
                 float* __restrict__ partials,
                 int nSamples)
{
  __shared__ float shP[TILE_FLOATS];
  __shared__ float shG[TILE_FLOATS];

  const int tid = threadIdx.x;
  const long long sampleBase = (long long)blockIdx.x * TILE;
  const long long floatBase  = sampleBase * FPS;
  long long rem = (long long)nSamples - sampleBase;
  const int samplesHere = (rem >= TILE) ? TILE : (int)rem;

  if (samplesHere == TILE) {
    // Full tile: 16B-aligned chunk base (TILE*FPS*4 = 32256 = 16*2016).
    const float* pSrc = pred + floatBase;
    const float* gSrc = gt   + floatBase;
    #pragma unroll 4
    for (int i = tid; i < TILE_VEC16; i += 256) {
      async_copy16(pSrc + i * 4, shP + i * 4);
      async_copy16(gSrc + i * 4, shG + i * 4);
    }
    wait_async0();
  } else if (samplesHere > 0) {
    const int nF = samplesHere * FPS;
    for (int i = tid; i < nF; i += 256) {
      shP[i] = pred[floatBase + i];
      shG[i] = gt[floatBase + i];
    }
  }
  __syncthreads();

  // Bone edge list (compile-time -> immediate LDS offsets after unroll).
  constexpr int BA[NBONES] = {3,2,20,8,9,10,20,4,5,6,20,1,0,16,17,18,0,12,13,14};
  constexpr int BB[NBONES] = {2,20,8,9,10,11,4,5,6,7,1,0,16,17,18,19,12,13,14,15};

  float acc = 0.0f;
  if (tid < samplesHere) {
    const float* jp = shP + tid * FPS;   // stride 63 across lanes: 63 ≡ -1 mod 64 banks -> conflict-free
    const float* jg = shG + tid * FPS;
    #pragma unroll
    for (int b = 0; b < NBONES; ++b) {
      const int a3 = BA[b] * 3, b3 = BB[b] * 3;
      float dx = jp[a3]     - jp[b3];
      float dy = jp[a3 + 1] - jp[b3 + 1];
      float dz = jp[a3 + 2] - jp[b3 + 2];
      float pl = sqrtf(dx * dx + dy * dy + dz * dz);
      float gx = jg[a3]     - jg[b3];
      float gy = jg[a3 + 1] - jg[b3 + 1];
      float gz = jg[a3 + 2] - jg[b3 + 2];
      float gl = sqrtf(gx * gx + gy * gy + gz * gz);
      acc += fabsf(pl - gl) / gl;
    }
  }

  // Deterministic block tree reduction (reuse shP after barrier).
  __syncthreads();
  shP[tid] = acc;
  __syncthreads();
  #pragma unroll
  for (int s = 128; s > 0; s >>= 1) {
    if (tid < s) shP[tid] += shP[tid + s];
    __syncthreads();
  }
  if (tid == 0) partials[blockIdx.x] = shP[0];
}

// ----- Kernel 2: deterministic final reduction ------------------------------
__global__ __launch_bounds__(256) void
final_reduce_kernel(const float* __restrict__ partials, int n,
                    float scale, float* __restrict__ out)
{
  __shared__ double sh[256];
  double s = 0.0;
  for (int i = threadIdx.x; i < n; i += 256) s += (double)partials[i];
  sh[threadIdx.x] = s;
  __syncthreads();
  #pragma unroll
  for (int k = 128; k > 0; k >>= 1) {
    if (threadIdx.x < k) sh[threadIdx.x] += sh[threadIdx.x + k];
    __syncthreads();
  }
  if (threadIdx.x == 0) out[0] = (float)(sh[0] * (double)scale);
}

// ----- Launch ---------------------------------------------------------------
extern "C" void kernel_launch(void* const* d_in, const int* in_sizes, int n_in,
                              void* d_out, int out_size, void* d_ws, size_t ws_size,
                              hipStream_t stream) {
  const float* pred = (const float*)d_in[0];
  const float* gt   = (const float*)d_in[1];
  const int nSamples = in_sizes[0] / FPS;          // 524288
  const int nTiles   = (nSamples + TILE - 1) / TILE;

  float* partials = (float*)d_ws;

  bone_tile_kernel<<<nTiles, 256, 0, stream>>>(pred, gt, partials, nSamples);

  const float scale = 1.0f / ((float)nSamples * (float)NBONES);
  final_reduce_kernel<<<1, 256, 0, stream>>>(partials, nTiles, scale,
                                             (float*)d_out);
}